// StratifiedSampling_FuzzyLayer_Scene15_89816356094166
// MI455X (gfx1250) — compile-verified
//
#include <hip/hip_runtime.h>

typedef float v2f __attribute__((ext_vector_type(2)));
typedef float v8f __attribute__((ext_vector_type(8)));

// Problem dimensions are fixed by the harness / reference:
//   x: [B, D] f32, pairs_idx: [B, P, 2] i32, rz/ry: [C] f32, out: [B, P*2, C] f32
constexpr int Bn   = 8192;
constexpr int Pn   = 60;
constexpr int Cn   = 15;
constexpr int Dn   = 200;
constexpr int ROWS = Bn * Pn * 2;           // 983040 output rows of length Cn

// Closed form of the 2-qubit circuit (product input state):
//   z0 = cos(ty)cos(x0) - sin(ty)cos(tz)sin(x0)sin(x1)
//   z1 = cos(ty)cos(x1)
//   out = (1+z)/2
// => out[rows x 15] = A[rows x 4] * W[4 x 15] + 0.5, one V_WMMA_F32_16X16X4_F32
//    per 16 output rows (wave32).
__global__ __launch_bounds__(256) void
StratifiedSampling_FuzzyLayer_Scene15_wmma_kernel(
    const float* __restrict__ x,
    const int*   __restrict__ pairs_idx,
    const float* __restrict__ rz_params,
    const float* __restrict__ ry_params,
    float*       __restrict__ out)
{
    const int lane = (int)(threadIdx.x & 31u);
    const int wave = (int)(threadIdx.x >> 5);
    const int r0   = (blockIdx.x * 8 + wave) * 16;   // 16 rows per wave, no tail

    const int half = lane >> 4;                       // 0: K0/K1 half, 1: K2/K3 half
    const int nm   = lane & 15;                       // A row index / B,D column index

    // ---------- B matrix (4 x 16 weights), 2 VGPRs ----------
    // VGPR0: lanes 0-15 -> K=0, lanes 16-31 -> K=2
    // VGPR1: lanes 0-15 -> K=1, lanes 16-31 -> K=3
    // W[0][c] = 0.5*cos(ty)   W[1][c] = -0.5*sin(ty)*cos(tz)
    // W[2][c] = 0.5*cos(ty)   W[3][c] = 0.5        (col 15 zero-padded)
    float w0 = 0.0f, w1 = 0.0f;
    if (nm < Cn) {
        const float ty = ry_params[nm];
        const float tz = rz_params[nm];
        w0 = 0.5f * __cosf(ty);                                   // K0 == K2
        w1 = half ? 0.5f : (-0.5f * __sinf(ty) * __cosf(tz));     // K3 : K1
    }
    v2f bmat; bmat.x = w0; bmat.y = w1;

    // ---------- A matrix (16 x 4 features), 2 VGPRs ----------
    // even rows (z0): [cos x0, sin x0*sin x1, 0, 1]
    // odd  rows (z1): [0, 0, cos x1, 1]
    const int r  = r0 + nm;
    const int bp = r >> 1;                // b*Pn + p
    const int j  = r & 1;                 // 0 -> z0 row, 1 -> z1 row
    const int b  = bp / Pn;

    // Both pair indices with one global_load_b64 (adjacent in memory).
    const int2  idx = ((const int2*)pairs_idx)[bp];
    const float x0  = x[b * Dn + idx.x];
    const float x1  = x[b * Dn + idx.y];
    float s0, c0, s1, c1;
    __sincosf(x0, &s0, &c0);
    __sincosf(x1, &s1, &c1);

    v2f amat;
    if (half == 0) {                      // K0, K1
        amat.x = j ? 0.0f : c0;
        amat.y = j ? 0.0f : (s0 * s1);
    } else {                              // K2, K3
        amat.x = j ? c1 : 0.0f;
        amat.y = 1.0f;
    }

    // ---------- D = A x B + 0.5 ----------
    v8f acc;
    #pragma unroll
    for (int i = 0; i < 8; ++i) acc[i] = 0.5f;
    acc = __builtin_amdgcn_wmma_f32_16x16x4_f32(
        /*neg_a=*/false, amat, /*neg_b=*/false, bmat,
        /*c_mod=*/(short)0, acc, /*reuse_a=*/false, /*reuse_b=*/false);

    // ---------- store: D VGPR i -> row (r0 + 8*half + i), column nm ----------
    // Base computed once; 8 stores with constant 60-byte immediate offsets.
    float* base = out + (size_t)((r0 + half * 8) * Cn + nm);
    if (nm < Cn) {
        #pragma unroll
        for (int i = 0; i < 8; ++i)
            base[i * Cn] = acc[i];
    }
}

extern "C" void kernel_launch(void* const* d_in, const int* in_sizes, int n_in,
                              void* d_out, int out_size, void* d_ws, size_t ws_size,
                              hipStream_t stream) {
    (void)in_sizes; (void)n_in; (void)out_size; (void)d_ws; (void)ws_size;
    const float* x     = (const float*)d_in[0];   // [B, D]
    const int*   pairs = (const int*)  d_in[1];   // [B, P, 2]
    const float* rz    = (const float*)d_in[2];   // [C]
    const float* ry    = (const float*)d_in[3];   // [C]
    float*       out   = (float*)d_out;           // [B, P*2, C]

    // ROWS = 983040 rows; 16 rows/wave, 8 waves/block -> 7680 blocks, no tail.
    const int blocks = ROWS / (16 * 8);
    StratifiedSampling_FuzzyLayer_Scene15_wmma_kernel
        <<<blocks, 256, 0, stream>>>(x, pairs, rz, ry, out);
}